// HungarianMatcher_23974507446520
// MI455X (gfx1250) — compile-verified
//
#include <hip/hip_runtime.h>
#include <hip/hip_bf16.h>

// Problem constants (match the reference).
#define BSZ 16
#define NQ  512   // columns m (queries)
#define NT  256   // rows n (targets)
#define COST_NODES 5.0f
#define COST_CLASS 3.0f

#define INF_F __builtin_huge_valf()

typedef __attribute__((ext_vector_type(2))) float v2f;
typedef __attribute__((ext_vector_type(8))) float v8f;

// ---------------------------------------------------------------------------
// CDNA5 async global->LDS copy (gfx1250). VDST = VGPR holding LDS byte addr,
// VADDR = 64-bit global address. Tracked by ASYNCcnt.
// ---------------------------------------------------------------------------
__device__ __forceinline__ void async_g2lds_b128(unsigned lds_addr, const void* gaddr) {
    asm volatile("global_load_async_to_lds_b128 %0, %1, off"
                 :: "v"(lds_addr), "v"(gaddr)
                 : "memory");
}

__device__ __forceinline__ void wait_asynccnt0() {
    asm volatile("s_wait_asynccnt 0" ::: "memory");
}

// Order-preserving float->uint map so (value, index) packs into one u64 min-key.
__device__ __forceinline__ unsigned long long ordkey(float f, int j) {
    unsigned b = __float_as_uint(f);
    b = (b & 0x80000000u) ? ~b : (b | 0x80000000u);
    return (((unsigned long long)b) << 32) | (unsigned)j;
}

// ---------------------------------------------------------------------------
// Kernel 1 (WMMA): cost(t,q) = 5 * L1(pred_q[0:3], tgt_t) - 3 * softmax(lg_q)[1]
// Exact identity: L1(x,y) = max_{s in {+-1}^3} [x0,x1,x2,1] . [s0,s1,s2,-(s.y)]
// => rank-4 GEMM on V_WMMA_F32_16X16X4_F32: A = 16 queries x 4,
//    B = 4 x 16 (2 targets x 8 sign patterns), then max over the 8 sign lanes.
// One block (16 waves) per batch; wave handles a 16-query block x 128 t-pairs.
// Output stored transposed as f16: costH[b][t][q].
// ---------------------------------------------------------------------------
__global__ __launch_bounds__(NQ, 1)
void hm_cost_wmma_kernel(const float* __restrict__ pred_nodes,   // (BSZ, NQ, 4)
                         const float* __restrict__ pred_logits,  // (BSZ, NQ, 2)
                         const float* __restrict__ tgt_nodes,    // (BSZ, NT, 3)
                         _Float16* __restrict__ costH) {         // (BSZ, NT, NQ)
    __shared__ float sx[NQ * 3];    // query coords
    __shared__ float sy[NT * 3];    // target coords
    __shared__ float pArr[NQ];      // softmax(logits)[1]

    const int b   = blockIdx.x;
    const int tid = threadIdx.x;

    {   // stage queries + class probability (thread = query)
        const float* src = pred_nodes + (size_t)b * NQ * 4;
        sx[tid * 3 + 0] = src[tid * 4 + 0];
        sx[tid * 3 + 1] = src[tid * 4 + 1];
        sx[tid * 3 + 2] = src[tid * 4 + 2];
        const float* lg = pred_logits + (size_t)b * NQ * 2;
        pArr[tid] = 1.0f / (1.0f + expf(lg[tid * 2 + 0] - lg[tid * 2 + 1]));
    }
    for (int i = tid; i < NT * 3; i += NQ) sy[i] = tgt_nodes[(size_t)b * NT * 3 + i];
    __syncthreads();

    const int lane = tid & 31;
    const int wave = tid >> 5;
    const int n    = lane & 15;     // D column within tile
    const int hi   = lane >> 4;     // lane half (A/B K-split, D M-split)
    // sign pattern for this lane's column: n = (t within pair)<<3 | s
    const float s0 = (n & 1) ? 1.f : -1.f;
    const float s1 = (n & 2) ? 1.f : -1.f;
    const float s2 = (n & 4) ? 1.f : -1.f;

    for (int qb = wave; qb < NQ / 16; qb += 16) {        // 2 query blocks / wave
        const int m0 = qb * 16;
        const int qA = m0 + n;                           // A: M = lane&15
        v2f a;                                           // 16x4 f32 A operand
        if (!hi) { a.x = sx[qA * 3 + 0]; a.y = sx[qA * 3 + 1]; }   // K=0,1
        else     { a.x = sx[qA * 3 + 2]; a.y = 1.0f; }             // K=2,3

        for (int tp = 0; tp < NT / 2; ++tp) {            // 128 target pairs
            const int t0 = tp * 2;
            const int t  = t0 + (n >> 3);
            v2f bb;                                      // 4x16 f32 B operand
            if (!hi) { bb.x = s0; bb.y = s1; }                     // K=0,1
            else {
                float bs = s0 * sy[t * 3 + 0] + s1 * sy[t * 3 + 1]
                         + s2 * sy[t * 3 + 2];
                bb.x = s2; bb.y = -bs;                             // K=2,3
            }
            v8f c = {};
            c = __builtin_amdgcn_wmma_f32_16x16x4_f32(
                    /*neg_a=*/false, a, /*neg_b=*/false, bb,
                    /*c_mod=*/(short)0, c, /*reuse_a=*/false, /*reuse_b=*/false);

            // max over the 8 sign patterns (lanes n, n^1, n^2, n^4)
            #pragma unroll
            for (int m = 1; m <= 4; m <<= 1) {
                #pragma unroll
                for (int r = 0; r < 8; ++r) {
                    float o = __shfl_xor(c[r], m, 32);
                    c[r] = fmaxf(c[r], o);
                }
            }
            if ((lane & 7) == 0) {                       // one writer per group
                const int qbase = m0 + hi * 8;           // D: M=r (+8 on hi half)
                _Float16* dst = costH + ((size_t)b * NT + t) * NQ + qbase;
                #pragma unroll
                for (int r = 0; r < 8; ++r)
                    dst[r] = (_Float16)(COST_NODES * c[r]
                                        - COST_CLASS * pArr[qbase + r]);
            }
        }
    }
}

// ---------------------------------------------------------------------------
// Kernel 2: per-batch LAPJV (scipy linear_sum_assignment recurrence), the
// whole 256x512 cost tile resident in LDS (f16, 256KB — needs CDNA5's 320KB
// WGP LDS), staged via async global->LDS loads. One workgroup per batch,
// 512 threads; thread j owns column j (v[j], SC[j], shortest[j] in regs).
// ---------------------------------------------------------------------------
// LDS layout (bytes):
#define OFF_COST   0            // _Float16[NT*NQ]            262144
#define OFF_SHORT  262144       // float[NQ]                    2048
#define OFF_PATH   264192       // int[NQ]                      2048
#define OFF_R4C    266240       // int[NQ]                      2048
#define OFF_C4R    268288       // int[NT]                      1024
#define OFF_U      269312       // float[NT]                    1024
#define OFF_SR     270336       // int[NT]                      1024
#define OFF_WAVE   271360       // u64[16]                       128
#define OFF_BCAST  271488       // int[8]                         32
#define LDS_BYTES  271520

__global__ __launch_bounds__(NQ, 1)
void hm_lapjv_kernel(const _Float16* __restrict__ costH,  // (BSZ, NT, NQ)
                     float* __restrict__ out) {           // (BSZ, 2, NT)
    extern __shared__ char smem[];
    _Float16* cost  = (_Float16*)(smem + OFF_COST);
    float*    sShort= (float*)   (smem + OFF_SHORT);
    int*      sPath = (int*)     (smem + OFF_PATH);
    int*      r4c   = (int*)     (smem + OFF_R4C);
    int*      c4r   = (int*)     (smem + OFF_C4R);
    float*    uArr  = (float*)   (smem + OFF_U);
    int*      sSR   = (int*)     (smem + OFF_SR);
    unsigned long long* sWave = (unsigned long long*)(smem + OFF_WAVE);
    volatile int* bcast = (int*)(smem + OFF_BCAST);

    const int b   = blockIdx.x;
    const int tid = threadIdx.x;

    // ---- async-stage this batch's 256KB cost tile HBM -> LDS -------------
    // Low 32 bits of a generic pointer into LDS are the wave-relative LDS
    // byte address (flat aperture mapping), which is what VDST expects.
    {
        const char* gbase = (const char*)(costH + (size_t)b * NT * NQ);
        unsigned     lbase = (unsigned)(unsigned long long)(void*)cost;
        #pragma unroll 4
        for (int k = 0; k < 32; ++k) {               // 512 thr * 32 * 16B = 256KB
            unsigned off = (unsigned)((k * NQ + tid) * 16);
            async_g2lds_b128(lbase + off, gbase + off);
        }
    }

    // ---- init solver state while the async copy is in flight -------------
    if (tid < NT) { uArr[tid] = 0.0f; c4r[tid] = -1; sSR[tid] = 0; }
    r4c[tid] = -1;
    sShort[tid] = INF_F;
    sPath[tid]  = -1;
    float v_j = 0.0f;                                // column potential (register)

    wait_asynccnt0();
    __syncthreads();

    for (int row = 0; row < NT; ++row) {
        float shortest_j = INF_F;                    // per-column shortest (reg)
        bool  SCj    = false;                        // column scanned flag (reg)
        int   i_cur  = row;
        float u_icur = uArr[row];
        float minval = 0.0f;
        int   sink   = -1;

        // ---- Dijkstra over columns; exact reference recurrence ----------
        while (true) {
            if (tid == 0) sSR[i_cur] = 1;            // SR mark (ref: top of body)

            float red = minval + (float)cost[i_cur * NQ + tid] - u_icur - v_j;
            if (!SCj && red < shortest_j) {
                shortest_j  = red;
                sShort[tid] = red;
                sPath[tid]  = i_cur;
            }
            float masked = SCj ? INF_F : shortest_j;

            // masked argmin over 512 columns: wave32 shuffle + 16-wave combine
            unsigned long long k = ordkey(masked, tid);
            #pragma unroll
            for (int o = 16; o > 0; o >>= 1) {
                unsigned long long other = __shfl_xor(k, o, 32);
                if (other < k) k = other;
            }
            if ((tid & 31) == 0) sWave[tid >> 5] = k;
            __syncthreads();
            if (tid == 0) {
                unsigned long long kmin = sWave[0];
                #pragma unroll
                for (int w = 1; w < 16; ++w)
                    if (sWave[w] < kmin) kmin = sWave[w];
                int jstar = (int)(kmin & 0xffffffffu);
                unsigned mb = (unsigned)(kmin >> 32);
                unsigned fb = (mb & 0x80000000u) ? (mb & 0x7fffffffu) : ~mb;
                bcast[0] = jstar;
                bcast[1] = r4c[jstar];
                bcast[2] = (int)fb;
            }
            __syncthreads();
            const int jstar = bcast[0];
            const int rj    = bcast[1];
            minval = __uint_as_float((unsigned)bcast[2]);

            if (tid == jstar) SCj = true;            // SC mark (ref: after argmin)
            if (rj < 0) { sink = jstar; break; }     // unmatched column -> done
            i_cur  = rj;
            u_icur = uArr[i_cur];
        }

        // ---- potential updates (pre-augmentation col4row, as in ref) ----
        if (SCj) v_j -= minval - shortest_j;         // v -= where(SC, minval-short)
        if (tid < NT) {
            if (tid == row) {
                uArr[tid] += minval;
            } else if (sSR[tid]) {
                int cc = c4r[tid]; if (cc < 0) cc = 0;
                uArr[tid] += minval - sShort[cc];
            }
        }
        __syncthreads();                             // u done before c4r mutates

        // ---- augmenting-path walk (single thread, <= 257 hops) ----------
        if (tid == 0) {
            int s = sink;
            while (s >= 0) {
                int i  = sPath[s];
                r4c[s] = i;
                int nxt = c4r[i];
                c4r[i]  = s;
                s = nxt;
            }
        }
        __syncthreads();

        // ---- reset per-row scratch --------------------------------------
        sShort[tid] = INF_F;
        sPath[tid]  = -1;
        if (tid < NT) sSR[tid] = 0;
        __syncthreads();
    }

    // ---- emit stack([qi, order]): counting-sort rank (columns distinct) --
    if (tid < NT) {
        int myc = c4r[tid];
        int rank = 0;
        for (int i = 0; i < NT; ++i) rank += (c4r[i] < myc) ? 1 : 0;
        out[(size_t)b * 2 * NT + rank]      = (float)myc;   // qi (sorted cols)
        out[(size_t)b * 2 * NT + NT + rank] = (float)tid;   // order (row ids)
    }
}

// ---------------------------------------------------------------------------
extern "C" void kernel_launch(void* const* d_in, const int* in_sizes, int n_in,
                              void* d_out, int out_size, void* d_ws, size_t ws_size,
                              hipStream_t stream) {
    (void)in_sizes; (void)n_in; (void)out_size; (void)ws_size;
    const float* pred_nodes  = (const float*)d_in[0];
    const float* pred_logits = (const float*)d_in[1];
    const float* tgt_nodes   = (const float*)d_in[2];
    float*       out         = (float*)d_out;
    _Float16*    costH       = (_Float16*)d_ws;      // 16*256*512*2 = 4 MB

    // Allow >64KB dynamic LDS (CDNA5 WGP: 320KB). Idempotent, deterministic.
    hipFuncSetAttribute((const void*)hm_lapjv_kernel,
                        hipFuncAttributeMaxDynamicSharedMemorySize, LDS_BYTES);

    hm_cost_wmma_kernel<<<dim3(BSZ), dim3(NQ), 0, stream>>>(pred_nodes, pred_logits,
                                                            tgt_nodes, costH);
    hm_lapjv_kernel<<<dim3(BSZ), dim3(NQ), LDS_BYTES, stream>>>(costH, out);
}